// expected_caibration_error_50242527428666
// MI455X (gfx1250) — compile-verified
//
#include <hip/hip_runtime.h>
#include <hip/hip_bf16.h>
#include <math.h>

#define NBINS  15
#define C      1000      // classes per row (fixed by reference)
#define ROWB   4000u     // bytes per row = C * 4
#define WAVES  8         // waves per 256-thread block (wave32)
#define FCHUNK 16        // finalize: fixed chunks per bin (determinism)

#if __has_builtin(__builtin_amdgcn_exp2f)
#define EXP2F(x) __builtin_amdgcn_exp2f(x)
#else
#define EXP2F(x) exp2f(x)
#endif

// Issue one row (1000 floats = 250 float4) as 8 async global->LDS b128 copies.
// Each instruction moves 32 lanes x 16B = 512B. Chunk 7 is lane-predicated
// (lanes 26..31 would be past element 1000) so we never read OOB in global.
// th:TH_LOAD_NT: rows are streamed once (262MB > 192MB L2) -> don't retain.
// ASYNCcnt += 8 per call (uniform across the wave: lanes 0..25 always active).
__device__ __forceinline__ void issue_row_async(const float* __restrict__ logits,
                                                unsigned rowByte, unsigned ldsByte,
                                                int lane) {
#pragma unroll
  for (int k = 0; k < 7; ++k) {
    unsigned goff  = rowByte + 16u * (unsigned)(lane + 32 * k);
    unsigned laddr = ldsByte + 16u * (unsigned)(lane + 32 * k);
    asm volatile("global_load_async_to_lds_b128 %0, %1, %2 th:TH_LOAD_NT"
                 :: "v"(laddr), "v"(goff), "s"(logits) : "memory");
  }
  if (lane < 26) {
    unsigned goff  = rowByte + 16u * (unsigned)(lane + 224);
    unsigned laddr = ldsByte + 16u * (unsigned)(lane + 224);
    asm volatile("global_load_async_to_lds_b128 %0, %1, %2 th:TH_LOAD_NT"
                 :: "v"(laddr), "v"(goff), "s"(logits) : "memory");
  }
}

// One wave processes one staged row from LDS: global max + first argmax +
// sum of exp(x - max). Returns confidence = 1/sumexp and prediction index
// (valid in all lanes).
__device__ __forceinline__ void process_row(const float* __restrict__ tile, int lane,
                                            float& conf, int& pred) {
  const float4* t4 = reinterpret_cast<const float4*>(tile);
  const float NINF = -__builtin_inff();
  float4 v[8];
#pragma unroll
  for (int k = 0; k < 7; ++k) v[k] = t4[lane + 32 * k];
  if (lane < 26) {
    v[7] = t4[lane + 224];
  } else {
    v[7].x = NINF; v[7].y = NINF; v[7].z = NINF; v[7].w = NINF;
  }

  // Per-lane max with first-occurrence index (in-lane indices are increasing).
  float bm = NINF; int bi = 0x7fffffff;
#pragma unroll
  for (int k = 0; k < 8; ++k) {
    const int base = 4 * (lane + 32 * k);
    if (v[k].x > bm) { bm = v[k].x; bi = base;     }
    if (v[k].y > bm) { bm = v[k].y; bi = base + 1; }
    if (v[k].z > bm) { bm = v[k].z; bi = base + 2; }
    if (v[k].w > bm) { bm = v[k].w; bi = base + 3; }
  }
  // Wave32 reduction; tie-break to the smallest index (matches jnp.argmax).
#pragma unroll
  for (int off = 16; off >= 1; off >>= 1) {
    const float om = __shfl_xor(bm, off, 32);
    const int   oi = __shfl_xor(bi, off, 32);
    if (om > bm || (om == bm && oi < bi)) { bm = om; bi = oi; }
  }

  // sum exp(x - m) via native v_exp_f32 (exp2). Masked -inf lanes give 0.
  const float L2E = 1.4426950408889634f;
  float s = 0.f;
#pragma unroll
  for (int k = 0; k < 8; ++k) {
    s += EXP2F((v[k].x - bm) * L2E);
    s += EXP2F((v[k].y - bm) * L2E);
    s += EXP2F((v[k].z - bm) * L2E);
    s += EXP2F((v[k].w - bm) * L2E);
  }
#pragma unroll
  for (int off = 16; off >= 1; off >>= 1) s += __shfl_xor(s, off, 32);

  conf = 1.0f / s;   // max softmax prob
  pred = bi;
}

__global__ __launch_bounds__(256) void ece_bins_kernel(
    const float* __restrict__ logits, const int* __restrict__ labels,
    float* __restrict__ partials, int N) {
  __shared__ float wbins[WAVES * 45];                  // per-wave {cnt,conf,acc}[15]
  __shared__ __align__(16) float tiles[2 * WAVES * C]; // 2 row slots per wave

  const int tid  = threadIdx.x;
  const int lane = tid & 31;
  const int w    = tid >> 5;

  for (int i = tid; i < WAVES * 45; i += 256) wbins[i] = 0.f;
  __syncthreads();

  // Low 32 bits of a generic pointer to LDS == LDS byte offset.
  const unsigned tilesBase = (unsigned)(unsigned long long)(const void*)tiles;
  float* tile0 = &tiles[(2 * w) * C];
  float* tile1 = &tiles[(2 * w + 1) * C];
  const unsigned lds0 = tilesBase + (unsigned)(2 * w) * ROWB;
  const unsigned lds1 = lds0 + ROWB;

  const int stride = (int)gridDim.x * WAVES;
  int r = (int)blockIdx.x * WAVES + w;
  int parity = 0;

  if (r < N) issue_row_async(logits, (unsigned)r * ROWB, lds0, lane);

  while (r < N) {
    const int rn = r + stride;
    // Ensure all LDS reads of the slot we are about to overwrite are done.
    asm volatile("s_wait_dscnt 0x0" ::: "memory");
    if (rn < N) {
      issue_row_async(logits, (unsigned)rn * ROWB, parity ? lds0 : lds1, lane);
      // 16 outstanding max; first 8 (current row) complete in order.
      asm volatile("s_wait_asynccnt 0x8" ::: "memory");
    } else {
      asm volatile("s_wait_asynccnt 0x0" ::: "memory");
    }

    float conf; int pred;
    process_row(parity ? tile1 : tile0, lane, conf, pred);

    if (lane == 0) {
      // bin = sum_{i=0..14}(conf > i/15) - 1 ; conf>0 always => valid.
      int b = 0;
#pragma unroll
      for (int i = 1; i < NBINS; ++i) b += (conf > (float)i / (float)NBINS) ? 1 : 0;
      const float av = (pred == labels[r]) ? 1.0f : 0.0f;
      wbins[w * 45 + b]      += 1.0f;
      wbins[w * 45 + 15 + b] += conf;
      wbins[w * 45 + 30 + b] += av;
    }
    parity ^= 1;
    r = rn;
  }

  __syncthreads();
  // Deterministic per-block fold of the 8 wave tables -> global partials.
  for (int i = tid; i < 45; i += 256) {
    float s = 0.f;
#pragma unroll
    for (int ww = 0; ww < WAVES; ++ww) s += wbins[ww * 45 + i];
    partials[(size_t)blockIdx.x * 45 + i] = s;
  }
}

// Deterministic two-level fold: 45 bins x FCHUNK fixed chunks of blocks,
// then 45 threads fold the chunk partials in fixed order. All summation
// orders are fixed by (bin, chunk) layout -> bit-reproducible every replay.
__global__ __launch_bounds__(45 * FCHUNK) void ece_finalize_kernel(
    const float* __restrict__ partials, float* __restrict__ out, int N, int nblk) {
  __shared__ float sh[FCHUNK * 45];
  __shared__ float tot[45];
  const int t = threadIdx.x;

  if (t < FCHUNK * 45) {
    const int bin   = t % 45;
    const int chunk = t / 45;
    const int step  = (nblk + FCHUNK - 1) / FCHUNK;
    const int b0    = chunk * step;
    int b1          = b0 + step;
    if (b1 > nblk) b1 = nblk;
    float s = 0.f;
    for (int b = b0; b < b1; ++b) s += partials[(size_t)b * 45 + bin];
    sh[chunk * 45 + bin] = s;
  }
  __syncthreads();

  if (t < 45) {
    float s = 0.f;
#pragma unroll
    for (int c = 0; c < FCHUNK; ++c) s += sh[c * 45 + t];
    tot[t] = s;
  }
  __syncthreads();

  if (t == 0) {
    float ece = 0.f, acc = 0.f;
    const float n = (float)N;
    for (int i = 0; i < NBINS; ++i) {
      const float c = tot[i];
      if (c > 0.f) {
        const float prop = c / n;
        const float ac = tot[15 + i] / c;  // avg confidence
        const float aa = tot[30 + i] / c;  // avg accuracy
        ece += fabsf(ac - aa) * prop;
        acc += aa * prop;
      }
    }
    out[0] = ece * 100.f;
    out[1] = acc * 100.f;
  }
}

extern "C" void kernel_launch(void* const* d_in, const int* in_sizes, int n_in,
                              void* d_out, int out_size, void* d_ws, size_t ws_size,
                              hipStream_t stream) {
  const float* logits = (const float*)d_in[0];
  const int*   labels = (const int*)d_in[1];
  float* out      = (float*)d_out;
  float* partials = (float*)d_ws;

  const int N = in_sizes[1];          // number of rows (labels count)

  int nblk = 1024;                                    // 8192 waves, 8 rows each
  const int maxByWs = (int)(ws_size / (45 * sizeof(float)));
  if (nblk > maxByWs) nblk = maxByWs;
  const int needed = (N + WAVES - 1) / WAVES;
  if (nblk > needed) nblk = needed;
  if (nblk < 1) nblk = 1;

  ece_bins_kernel<<<nblk, 256, 0, stream>>>(logits, labels, partials, N);
  ece_finalize_kernel<<<1, 45 * FCHUNK, 0, stream>>>(partials, out, N, nblk);
}